// PytorchCreator_58506044506192
// MI455X (gfx1250) — compile-verified
//
#include <hip/hip_runtime.h>

// ---------------------------------------------------------------------------
// L1=5 layers, N=64 nodes, K=8 parents/node, D=128.
//   acc[n,e] = sum_k sum_d prev[idx[n,k], d] * W[n,k,d,e];  out = logistic(acc+b)
// Pure weight-stream bound: 168 MB fp32 read once (fits the 192 MB L2 for
// graph-replay steady state). Strategy: fetch weight rows as contiguous
// global_load_b128 (512 B per wave-instruction), stage through a wave-private
// LDS slab, build V_WMMA_F32_16X16X4_F32 B-fragments from LDS, activation
// vector replicated over the 16 M rows of A.
// ---------------------------------------------------------------------------

typedef float v2f __attribute__((ext_vector_type(2)));
typedef float v8f __attribute__((ext_vector_type(8)));

#define N_NODES 64
#define KFAN    8
#define DIM     128
#define NLAYER  5
#define ECOLS   64        // output columns handled per workgroup (4 WMMA tiles)
#define TROWS   8         // weight rows staged per block
#define NBLK    (DIM / TROWS)

__device__ __forceinline__ float logisticf(float x) {
    return 1.0f / (1.0f + expf(-x));
}

// Layer 0: elementwise logistic on the external inputs.
__global__ void act0_kernel(const float* __restrict__ in,
                            float* __restrict__ out, int n) {
    int i = blockIdx.x * blockDim.x + threadIdx.x;
    if (i < n) out[i] = logisticf(in[i]);
}

// One layer. Grid: 128 blocks = (node n, 64-col half eh).
// Block: 256 threads = 8 waves; wave w handles parent k = w, all 4 col-tiles.
__global__ __launch_bounds__(256, 2)
void layer_kernel(const float* __restrict__ prev,   // [64][128]
                  const float* __restrict__ W,      // [64][8][128][128] (layer)
                  const float* __restrict__ bias,   // [64][128]         (layer)
                  const int*   __restrict__ pidx,   // [64][8]           (layer)
                  float*       __restrict__ out)    // [64][128]
{
    __shared__ float vsh[KFAN][DIM];            // parent activation vectors
    __shared__ float wbuf[KFAN][TROWS * ECOLS]; // wave-private weight slabs
    __shared__ float partial[KFAN][ECOLS];      // per-wave partial results

    const int bx    = blockIdx.x;
    const int n     = bx >> 1;                  // node
    const int eh    = bx & 1;                   // column half
    const int ecol0 = eh * ECOLS;

    const int tid  = threadIdx.x;
    const int wave = tid >> 5;                  // == parent index k
    const int lane = tid & 31;

    // Stage this wave's parent vector into its LDS slot (wave-private:
    // same-wave LDS ops are in-order, no barrier needed before use).
    const int parent = pidx[n * KFAN + wave];
    const float* pv  = prev + (size_t)parent * DIM;
    vsh[wave][lane]      = pv[lane];
    vsh[wave][lane + 32] = pv[lane + 32];
    vsh[wave][lane + 64] = pv[lane + 64];
    vsh[wave][lane + 96] = pv[lane + 96];

    // WMMA fragment addressing (wave32, f32 16x16x4):
    //  A 16x4 : lane L -> a.x = v[d0+rowoff], a.y = v[d0+rowoff+1]
    //  B 4x16 : lane L -> b.x = B[K=rowoff][N=col], b.y = B[rowoff+1][col]
    const int col    = lane & 15;
    const int rowoff = (lane < 16) ? 0 : 2;

    // Staging lane map: each lane loads 16 B; one b128 instruction covers
    // two full 256 B row-segments (rows 2 lanes>>4 apart).
    const int srow = lane >> 4;                 // 0..1
    const int scol = (lane & 15) * 4;           // 0..60

    const float* Wk = W + (size_t)(n * KFAN + wave) * DIM * DIM;
    float* slab = wbuf[wave];

    v8f c0 = {}, c1 = {}, c2 = {}, c3 = {};

    for (int j = 0; j < NBLK; ++j) {
        const int dbase = j * TROWS;

        // ---- stage TROWS x ECOLS weight block: 4 x global_load_b128 ----
#pragma unroll
        for (int i = 0; i < 4; ++i) {
            const int r = i * 2 + srow;         // row within block
            const float4 t = *reinterpret_cast<const float4*>(
                Wk + (size_t)(dbase + r) * DIM + ecol0 + scol);
            *reinterpret_cast<float4*>(&slab[r * ECOLS + scol]) = t;
        }

        // ---- consume: 2 d-chunks x 4 col-tiles = 8 WMMAs per block ----
#pragma unroll
        for (int dd = 0; dd < TROWS; dd += 4) {
            v2f a;
            a.x = vsh[wave][dbase + dd + rowoff];      // -> ds_load_b64
            a.y = vsh[wave][dbase + dd + rowoff + 1];
            const float* s0 = &slab[(dd + rowoff) * ECOLS + col];
            const float* s1 = s0 + ECOLS;              // next weight row
            v2f b;
            b.x = s0[0];  b.y = s1[0];
            c0 = __builtin_amdgcn_wmma_f32_16x16x4_f32(false, a, false, b,
                                                       (short)0, c0, false, false);
            b.x = s0[16]; b.y = s1[16];
            c1 = __builtin_amdgcn_wmma_f32_16x16x4_f32(false, a, false, b,
                                                       (short)0, c1, false, false);
            b.x = s0[32]; b.y = s1[32];
            c2 = __builtin_amdgcn_wmma_f32_16x16x4_f32(false, a, false, b,
                                                       (short)0, c2, false, false);
            b.x = s0[48]; b.y = s1[48];
            c3 = __builtin_amdgcn_wmma_f32_16x16x4_f32(false, a, false, b,
                                                       (short)0, c3, false, false);
        }
    }

    // D layout: lanes 0-15, VGPR0 hold row M=0 (all M rows identical here).
    if (lane < 16) {
        partial[wave][col]      = c0[0];
        partial[wave][col + 16] = c1[0];
        partial[wave][col + 32] = c2[0];
        partial[wave][col + 48] = c3[0];
    }
    __syncthreads();

    // Deterministic fixed-order reduction over the 8 parents + bias + logistic.
    if (tid < ECOLS) {
        const int e = ecol0 + tid;
        float s = bias[n * DIM + e];
#pragma unroll
        for (int w = 0; w < KFAN; ++w) s += partial[w][tid];
        out[n * DIM + e] = logisticf(s);
    }
}

extern "C" void kernel_launch(void* const* d_in, const int* in_sizes, int n_in,
                              void* d_out, int out_size, void* d_ws, size_t ws_size,
                              hipStream_t stream) {
    (void)in_sizes; (void)n_in; (void)out_size; (void)ws_size;

    const float* inputs  = (const float*)d_in[0];  // [64,128]
    const float* weights = (const float*)d_in[1];  // [5,64,8,128,128]
    const float* biases  = (const float*)d_in[2];  // [5,64,128]
    const int*   pidx    = (const int*)d_in[3];    // [5,64,8]

    float* out  = (float*)d_out;
    float* bufA = (float*)d_ws;    // 32 KB scratch
    float* bufB = out;             // reuse d_out as the second ping-pong buffer

    const int NE = N_NODES * DIM;  // 8192
    act0_kernel<<<(NE + 255) / 256, 256, 0, stream>>>(inputs, bufA, NE);

    const size_t wlayer = (size_t)N_NODES * KFAN * DIM * DIM;
    const float* cur = bufA;
    for (int l = 0; l < NLAYER; ++l) {
        float* dst = (l == NLAYER - 1) ? out : ((l & 1) ? bufA : bufB);
        layer_kernel<<<N_NODES * 2, 256, 0, stream>>>(
            cur,
            weights + (size_t)l * wlayer,
            biases  + (size_t)l * NE,
            pidx    + (size_t)l * N_NODES * KFAN,
            dst);
        cur = dst;
    }
}